// FNO_U_84533546320187
// MI455X (gfx1250) — compile-verified
//
#include <hip/hip_runtime.h>
#include <cmath>

typedef __attribute__((ext_vector_type(2))) float v2f;
typedef __attribute__((ext_vector_type(8))) float v8f;

#define PTS 8  // spatial points per workgroup == waves per workgroup

__device__ __forceinline__ float gelu_erf(float x) {
    return 0.5f * x * (1.0f + erff(x * 0.70710678118654752440f));
}

// One r_trans stage: Y[b, o, p] = gelu( sum_i A[b,i,p] * W[i,o,p] + Bias[o,p] )
// A is gathered from inX (ciX chans) ++ inK (ciK chans) with low/high "corner"
// row mapping; output is written (or corner-scatter-added) into `out`.
// M=16 (batch) x N=CO x K=ciTot complex matmul per point via f32 WMMA 16x16x4.
template <int NT>
__global__ __launch_bounds__(256, 1) void fno_rtrans(
    const float* __restrict__ inX, int ciX, int S1x, int M2x,
    const float* __restrict__ inK, int ciK, int S1k, int M2k,
    const float* __restrict__ W,  const float* __restrict__ Bias,
    float* __restrict__ out, int S1t, int M2t, int accumulate,
    int m1, int m2)
{
    constexpr int CO = NT * 16;
    constexpr int LOG2CO = (NT == 2) ? 5 : (NT == 4) ? 6 : 7;
    constexpr int BROW = CO * 4 + 4;   // +4 floats pad -> LDS store stride % 64 != 0

    __shared__ __align__(16) float  sAre[PTS * 68];      // [p][b(16)][k(4)] pad 68
    __shared__ __align__(16) float  sAim[PTS * 68];
    __shared__ __align__(16) float  sBre[PTS * BROW];    // [p][o(CO)][k(4)]
    __shared__ __align__(16) float  sBim[PTS * BROW];
    __shared__ __align__(16) float2 sBias[PTS * CO];     // [p][o]
    __shared__ __align__(16) float2 sOut[16 * 16 * PTS]; // [b][o_local][p]

    const int tid  = threadIdx.x;
    const int lane = tid & 31;
    const int wave = tid >> 5;              // local point index 0..7
    const int p0   = blockIdx.x * PTS;
    const int ciTot = ciX + ciK;
    const int nptsW = 2 * m1 * m2;          // weight/bias plane size

    // ---- stage bias into LDS (coalesced in p) ----
    for (int e = tid; e < CO * PTS; e += 256) {
        int p = e & (PTS - 1);
        int o = e >> 3;
        sBias[p * CO + o] = *(const float2*)(Bias + 2 * ((size_t)o * nptsW + (p0 + p)));
    }

    v8f accR[NT], accI[NT];
#pragma unroll
    for (int t = 0; t < NT; ++t) {
        accR[t] = (v8f){0.f,0.f,0.f,0.f,0.f,0.f,0.f,0.f};
        accI[t] = (v8f){0.f,0.f,0.f,0.f,0.f,0.f,0.f,0.f};
    }

    const int mrow = lane & 15;             // A: M row / B: N col / D: N col
    const int kp   = (lane >> 4) * 2;       // K sub-pair per lane half

    for (int kc = 0; kc < ciTot; kc += 4) {
        __syncthreads();
        // ---- A chunk: X[b, kc..kc+3, p] -> LDS (corner-gathered) ----
        for (int e = tid; e < 16 * 4 * PTS; e += 256) {
            int p = e & (PTS - 1);
            int bk = e >> 3;
            int b = bk & 15;
            int k = bk >> 4;
            int i = kc + k;
            int pg = p0 + p;
            int r = pg / m2, c = pg - r * m2;
            const float* src;
            if (i < ciX) {
                int rr = (r < m1) ? r : (S1x - 2 * m1 + r);
                src = inX + 2 * (((size_t)(b * ciX + i) * S1x + rr) * M2x + c);
            } else {
                int j = i - ciX;
                int rr = (r < m1) ? r : (S1k - 2 * m1 + r);
                src = inK + 2 * (((size_t)(b * ciK + j) * S1k + rr) * M2k + c);
            }
            float2 v = *(const float2*)src;
            sAre[p * 68 + b * 4 + k] = v.x;
            sAim[p * 68 + b * 4 + k] = v.y;
        }
        // ---- B chunk: W[kc..kc+3, 0..CO-1, p] -> LDS (coalesced in p) ----
        for (int e = tid; e < 4 * CO * PTS; e += 256) {
            int p = e & (PTS - 1);
            int rest = e >> 3;
            int o = rest & (CO - 1);
            int k = rest >> LOG2CO;
            int i = kc + k;
            float2 v = *(const float2*)(W + 2 * ((size_t)(i * CO + o) * nptsW + (p0 + p)));
            sBre[p * BROW + o * 4 + k] = v.x;
            sBim[p * BROW + o * 4 + k] = v.y;
        }
        __syncthreads();
        // ---- complex WMMA: Cr += ArBr - AiBi ; Ci += ArBi + AiBr ----
        v2f aR = *(const v2f*)&sAre[wave * 68 + mrow * 4 + kp];
        v2f aI = *(const v2f*)&sAim[wave * 68 + mrow * 4 + kp];
        v2f aIn = -aI;   // f32 WMMA has no A/B NEG modifier; negate in VALU
#pragma unroll
        for (int t = 0; t < NT; ++t) {
            int o = t * 16 + mrow;
            v2f bR = *(const v2f*)&sBre[wave * BROW + o * 4 + kp];
            v2f bI = *(const v2f*)&sBim[wave * BROW + o * 4 + kp];
            accR[t] = __builtin_amdgcn_wmma_f32_16x16x4_f32(false, aR,  false, bR, (short)0, accR[t], false, false);
            accR[t] = __builtin_amdgcn_wmma_f32_16x16x4_f32(false, aIn, false, bI, (short)0, accR[t], false, false);
            accI[t] = __builtin_amdgcn_wmma_f32_16x16x4_f32(false, aR,  false, bI, (short)0, accI[t], false, false);
            accI[t] = __builtin_amdgcn_wmma_f32_16x16x4_f32(false, aI,  false, bR, (short)0, accI[t], false, false);
        }
    }

    // ---- epilogue: bias + erf-gelu (re/im separately), staged coalesced store ----
    const int mhi = (lane >> 4) * 8;
#pragma unroll
    for (int t = 0; t < NT; ++t) {
        float2 bb = sBias[wave * CO + t * 16 + mrow];
#pragma unroll
        for (int v = 0; v < 8; ++v) {
            float2 q;
            q.x = gelu_erf(accR[t][v] + bb.x);
            q.y = gelu_erf(accI[t][v] + bb.y);
            int m = mhi + v;
            sOut[(m * 16 + mrow) * PTS + wave] = q;
        }
        __syncthreads();
        for (int e = tid; e < 16 * 16 * PTS; e += 256) {
            int p = e & (PTS - 1);
            int rest = e >> 3;
            int o = t * 16 + (rest & 15);
            int b = rest >> 4;
            int pg = p0 + p;
            int r = pg / m2, c = pg - r * m2;
            int rr = (r < m1) ? r : (S1t - 2 * m1 + r);   // identity when S1t==2*m1
            float* dst = out + 2 * (((size_t)(b * CO + o) * S1t + rr) * M2t + c);
            float2 q = sOut[e];
            if (accumulate) {
                float2 old = *(const float2*)dst;
                q.x += old.x; q.y += old.y;
            }
            *(float2*)dst = q;
        }
        __syncthreads();
    }
}

extern "C" void kernel_launch(void* const* d_in, const int* in_sizes, int n_in,
                              void* d_out, int out_size, void* d_ws, size_t ws_size,
                              hipStream_t stream) {
    (void)in_sizes; (void)n_in; (void)out_size; (void)ws_size;
    const float* x     = (const float*)d_in[0];   // [16,32,128,65,2]
    const float* kfeat = (const float*)d_in[1];   // [16,12,128,65,2]
    const float* w0  = (const float*)d_in[2];
    const float* b0  = (const float*)d_in[3];
    const float* wd1 = (const float*)d_in[4];
    const float* bd1 = (const float*)d_in[5];
    const float* wu1 = (const float*)d_in[6];
    const float* bu1 = (const float*)d_in[7];
    const float* wd2 = (const float*)d_in[8];
    const float* bd2 = (const float*)d_in[9];
    const float* wu2 = (const float*)d_in[10];
    const float* bu2 = (const float*)d_in[11];

    float* out0 = (float*)d_out;                        // [16,32,128,65] cplx
    float* out1 = (float*)d_ws;                         // [16,64,64,32]  cplx
    float* out2 = out1 + (size_t)16 * 64 * 64 * 32 * 2; // [16,128,32,16] cplx

    dim3 blk(256);
    // level 0: ci = 32+12 -> co = 32, plane 128x65 (corner map is identity)
    fno_rtrans<2><<<8320 / PTS, blk, 0, stream>>>(x, 32, 128, 65, kfeat, 12, 128, 65,
                                                  w0, b0, out0, 128, 65, 0, 64, 65);
    // down level 1: corners(out0,32,32) ++ corners(kfeat) -> co = 64, plane 64x32
    fno_rtrans<4><<<2048 / PTS, blk, 0, stream>>>(out0, 32, 128, 65, kfeat, 12, 128, 65,
                                                  wd1, bd1, out1, 64, 32, 0, 32, 32);
    // down level 2: corners(out1,16,16) ++ corners(kfeat) -> co = 128, plane 32x16
    fno_rtrans<8><<<512 / PTS, blk, 0, stream>>>(out1, 64, 64, 32, kfeat, 12, 128, 65,
                                                 wd2, bd2, out2, 32, 16, 0, 16, 16);
    // up level 2: full out2 -> co = 64, corner-scatter-add into out1
    fno_rtrans<4><<<512 / PTS, blk, 0, stream>>>(out2, 128, 32, 16, nullptr, 0, 1, 1,
                                                 wu2, bu2, out1, 64, 32, 1, 16, 16);
    // up level 1: full (updated) out1 -> co = 32, corner-scatter-add into out0
    fno_rtrans<2><<<2048 / PTS, blk, 0, stream>>>(out1, 64, 64, 32, nullptr, 0, 1, 1,
                                                  wu1, bu1, out0, 128, 65, 1, 32, 32);
}